// ECELossBinary_76132590289249
// MI455X (gfx1250) — compile-verified
//
#include <hip/hip_runtime.h>

// ECE (binary) for N float32 confidences + int32 {0,1} targets.
// HBM-bound: 268 MB @ 23.3 TB/s => ~11.5 us floor. Strategy:
//   zero ws -> streaming half-wave-private LDS histogram -> tiny epilogue.
// Half-wave privatization puts the two 16-lane groups of each wave32 on
// disjoint LDS bank sets (u64 region = 128 B = banks 0-31 / 32-63), so a
// single ds_add sees ~1 expected same-address conflict instead of ~2+.

#define ECE_NBINS   15
#define ECE_NSEG    16          // 15 bins + dump bin for conf<=0
#define ECE_THREADS 256         // 8 wave32 per block
#define ECE_NSUB    (ECE_THREADS / 16)  // 16 half-wave histogram regions
#define ECE_BLOCKS  2048        // 16K waves in flight

// ws layout (floats): [0..15] counts, [16..31] conf_sum, [32..47] acc_sum
__global__ void ece_zero_kernel(float* __restrict__ ws) {
    int t = threadIdx.x;
    if (t < 3 * ECE_NSEG) ws[t] = 0.0f;
}

__device__ __forceinline__ int ece_bin(float c) {
    // reference: idx = clip(ceil(c*15)-1, 0, 14); conf<=0 -> dump bin 15
    int b = (int)__builtin_ceilf(c * 15.0f) - 1;
    b = b < 0 ? 0 : (b > (ECE_NBINS - 1) ? (ECE_NBINS - 1) : b);
    return (c > 0.0f) ? b : ECE_NBINS;
}

__global__ void __launch_bounds__(ECE_THREADS)
ece_hist_kernel(const float* __restrict__ conf,
                const int*   __restrict__ tgt,
                float*       __restrict__ ws,
                int n) {
    // Half-wave-private LDS histograms: count+acc packed in u64 (ds_add_u64),
    // conf sum as f32 (ds_add_f32). 2 LDS atomics per element.
    __shared__ unsigned long long s_ca[ECE_NSUB * ECE_NSEG];
    __shared__ float              s_cf[ECE_NSUB * ECE_NSEG];

    const int tid = threadIdx.x;
    const int sub = tid >> 4;   // half-wave id, 0..15

    for (int i = tid; i < ECE_NSUB * ECE_NSEG; i += ECE_THREADS) {
        s_ca[i] = 0ull;
        s_cf[i] = 0.0f;
    }
    __syncthreads();

    unsigned long long* my_ca = &s_ca[sub * ECE_NSEG];
    float*              my_cf = &s_cf[sub * ECE_NSEG];

    const int gid    = blockIdx.x * ECE_THREADS + tid;
    const int stride = gridDim.x * ECE_THREADS;
    const int nvec   = n >> 2;

    const float4* __restrict__ c4 = (const float4*)conf;
    const int4*   __restrict__ t4 = (const int4*)tgt;

#define ECE_PROC(cv, tv)                                                   \
    do {                                                                   \
        float c_ = (cv);                                                   \
        int   b_ = ece_bin(c_);                                            \
        int   p_ = (c_ >= 0.5f) ? 1 : 0;                                   \
        unsigned long long inc_ =                                          \
            1ull | ((p_ == (tv)) ? (1ull << 32) : 0ull);                   \
        atomicAdd(&my_ca[b_], inc_);  /* ds_add_u64 */                     \
        atomicAdd(&my_cf[b_], c_);    /* ds_add_f32 */                     \
    } while (0)

    for (int i = gid; i < nvec; i += stride) {
        // 128-bit vector loads (global_load_b128); speculative prefetch of
        // the next grid-stride iteration (safe out-of-range: TH=0 prefetch
        // is dropped on translation failure per ISA).
        __builtin_prefetch((const char*)&c4[i + stride], 0, 0);
        __builtin_prefetch((const char*)&t4[i + stride], 0, 0);
        float4 c = c4[i];
        int4   t = t4[i];
        ECE_PROC(c.x, t.x);
        ECE_PROC(c.y, t.y);
        ECE_PROC(c.z, t.z);
        ECE_PROC(c.w, t.w);
    }
    // scalar tail for n % 4
    for (int i = (nvec << 2) + gid; i < n; i += stride) {
        ECE_PROC(conf[i], tgt[i]);
    }
#undef ECE_PROC

    __syncthreads();

    // Reduce the 16 half-wave histograms, one bin per lane, then merge
    // globally (48 f32 atomics per block; 98K total over 48 addresses --
    // negligible vs the 11.5 us memory floor).
    if (tid < ECE_NSEG) {
        unsigned long long ca = 0ull;
        float cf = 0.0f;
        for (int s = 0; s < ECE_NSUB; ++s) {
            ca += s_ca[s * ECE_NSEG + tid];
            cf += s_cf[s * ECE_NSEG + tid];
        }
        float cnt = (float)(unsigned int)(ca & 0xffffffffull);
        float acc = (float)(unsigned int)(ca >> 32);
        atomicAdd(&ws[tid],                cnt); // global_atomic_add_f32
        atomicAdd(&ws[ECE_NSEG + tid],     cf);
        atomicAdd(&ws[2 * ECE_NSEG + tid], acc);
    }
}

__global__ void ece_final_kernel(const float* __restrict__ ws,
                                 float* __restrict__ out, int n) {
    __shared__ float terms[ECE_NBINS];
    int b = threadIdx.x;
    if (b < ECE_NBINS) {
        float cnt   = ws[b];
        float csum  = ws[ECE_NSEG + b];
        float asum  = ws[2 * ECE_NSEG + b];
        float denom = fmaxf(cnt, 1.0f);
        float gap   = fabsf(csum / denom - asum / denom);
        float prop  = cnt / (float)n;
        terms[b] = (cnt > 0.0f) ? gap * prop : 0.0f;
    }
    __syncthreads();
    if (b == 0) {
        float s = 0.0f;
        for (int i = 0; i < ECE_NBINS; ++i) s += terms[i];
        out[0] = s;
    }
}

extern "C" void kernel_launch(void* const* d_in, const int* in_sizes, int n_in,
                              void* d_out, int out_size, void* d_ws, size_t ws_size,
                              hipStream_t stream) {
    const float* conf = (const float*)d_in[0];
    const int*   tgt  = (const int*)d_in[1];
    float*       ws   = (float*)d_ws;
    float*       out  = (float*)d_out;
    const int n = in_sizes[0];

    ece_zero_kernel<<<1, 64, 0, stream>>>(ws);
    ece_hist_kernel<<<ECE_BLOCKS, ECE_THREADS, 0, stream>>>(conf, tgt, ws, n);
    ece_final_kernel<<<1, 32, 0, stream>>>(ws, out, n);
}